// MyGCNConv_37074157699598
// MI455X (gfx1250) — compile-verified
//
#include <hip/hip_runtime.h>

typedef float v2f __attribute__((ext_vector_type(2)));
typedef float v8f __attribute__((ext_vector_type(8)));

#define DIN 256
#define DOUT 128

// ---------------------------------------------------------------------------
// Kernel 1: h[n, 128] = x[n, 256] @ W[128, 256]^T  using V_WMMA_F32_16X16X4_F32
// One wave computes a 16-row x 128-col strip (8 column tiles of 16).
// A-frag layout (16x4 f32): lane<16 -> row=lane, K={k,k+1}; lane>=16 -> K={k+2,k+3}
// B-frag layout (4x16 f32): lane<16 -> col=lane, K={k,k+1}; lane>=16 -> K={k+2,k+3}
//   B[k][n] = W[n][k] -> two consecutive floats of row n of W.
// ---------------------------------------------------------------------------
__global__ __launch_bounds__(256) void gcn_gemm_wmma(const float* __restrict__ x,
                                                     const float* __restrict__ W,
                                                     float* __restrict__ h,
                                                     int n, int mtiles) {
    const int tile = blockIdx.x * (blockDim.x >> 5) + (threadIdx.x >> 5);
    if (tile >= mtiles) return;                 // wave-uniform exit, EXEC stays full
    const int lane = threadIdx.x & 31;
    const int half = lane >> 4;                 // 0: K lo pair, 1: K hi pair
    const int l16  = lane & 15;

    int arow = tile * 16 + l16;
    if (arow >= n) arow = n - 1;                // clamp so WMMA keeps EXEC all-1s
    const float* ap = x + (size_t)arow * DIN + half * 2;

    v8f acc[8];
#pragma unroll
    for (int ct = 0; ct < 8; ++ct) acc[ct] = (v8f)(0.0f);

    for (int k = 0; k < DIN; k += 4) {
        v2f a = *(const v2f*)(ap + k);
#pragma unroll
        for (int ct = 0; ct < 8; ++ct) {
            const float* bp = W + (size_t)(ct * 16 + l16) * DIN + half * 2 + k;
            v2f b = *(const v2f*)bp;
            // (neg_a, A, neg_b, B, c_mod, C, reuse_a, reuse_b)
            acc[ct] = __builtin_amdgcn_wmma_f32_16x16x4_f32(
                false, a, false, b, (short)0, acc[ct], false, false);
        }
    }

    // D layout: VGPR j: lane<16 -> (M=j, N=lane); lane>=16 -> (M=8+j, N=lane-16)
    const int rbase = tile * 16 + half * 8;
#pragma unroll
    for (int j = 0; j < 8; ++j) {
        const int r = rbase + j;
        if (r < n) {
#pragma unroll
            for (int ct = 0; ct < 8; ++ct)
                h[(size_t)r * DOUT + ct * 16 + l16] = acc[ct][j];
        }
    }
}

// ---------------------------------------------------------------------------
// Kernel 2: out[i] = (1/deg(i)) * sum_{e in row i} h[idx[e]]
// One wave per destination node; lane owns float4 -> 32*4 = 128 channels.
// h (51.2 MB) is L2-resident on MI455X (192 MB L2), so gathers hit L2.
// ---------------------------------------------------------------------------
__global__ __launch_bounds__(256) void gcn_spmm(const float* __restrict__ h,
                                                const int* __restrict__ ptr,
                                                const int* __restrict__ idx,
                                                float* __restrict__ out, int n) {
    const int node = blockIdx.x * (blockDim.x >> 5) + (threadIdx.x >> 5);
    if (node >= n) return;
    const int lane = threadIdx.x & 31;

    const int s = ptr[node];
    const int e = ptr[node + 1];

    float ax = 0.f, ay = 0.f, az = 0.f, aw = 0.f;
    int j = s;
    for (; j + 1 < e; j += 2) {
        const int s0 = idx[j];
        const int s1 = idx[j + 1];
        const float4 v0 = ((const float4*)(h + (size_t)s0 * DOUT))[lane];
        const float4 v1 = ((const float4*)(h + (size_t)s1 * DOUT))[lane];
        ax += v0.x + v1.x;  ay += v0.y + v1.y;
        az += v0.z + v1.z;  aw += v0.w + v1.w;
    }
    if (j < e) {
        const int s0 = idx[j];
        const float4 v0 = ((const float4*)(h + (size_t)s0 * DOUT))[lane];
        ax += v0.x;  ay += v0.y;  az += v0.z;  aw += v0.w;
    }

    const float inv = (e > s) ? 1.0f / (float)(e - s) : 0.0f;
    float4 r;
    r.x = ax * inv;  r.y = ay * inv;  r.z = az * inv;  r.w = aw * inv;
    ((float4*)(out + (size_t)node * DOUT))[lane] = r;
}

// ---------------------------------------------------------------------------
// inputs: [0]=x (N*256 f32), [1]=W (128*256 f32), [2]=ptr (N+1 int),
//         [3]=idx (E int), [4]=dst (E int, unused: CSR row == dst)
// d_ws: h scratch, N*128*4 = 51.2 MB
// ---------------------------------------------------------------------------
extern "C" void kernel_launch(void* const* d_in, const int* in_sizes, int n_in,
                              void* d_out, int out_size, void* d_ws, size_t ws_size,
                              hipStream_t stream) {
    const float* x   = (const float*)d_in[0];
    const float* W   = (const float*)d_in[1];
    const int*   ptr = (const int*)d_in[2];
    const int*   idx = (const int*)d_in[3];
    (void)n_in; (void)out_size; (void)ws_size;

    const int n = in_sizes[2] - 1;          // ptr has N+1 entries
    float* h   = (float*)d_ws;
    float* out = (float*)d_out;

    const int mtiles = (n + 15) / 16;
    const int wavesPerBlock = 8;            // 256 threads = 8 wave32

    dim3 blk(256);
    dim3 g1((mtiles + wavesPerBlock - 1) / wavesPerBlock);
    gcn_gemm_wmma<<<g1, blk, 0, stream>>>(x, W, h, n, mtiles);

    dim3 g2((n + wavesPerBlock - 1) / wavesPerBlock);
    gcn_spmm<<<g2, blk, 0, stream>>>(h, ptr, idx, out, n);
}